// GNN_LSTM_Regressor_5746666242542
// MI455X (gfx1250) — compile-verified
//
#include <hip/hip_runtime.h>

typedef __attribute__((ext_vector_type(2))) float v2f;
typedef __attribute__((ext_vector_type(8))) float v8f;

#define NN   10000
#define BNN  40000
#define EE   320000
#define HID  64
#define WW   12

// ---------------- degree / normalization ----------------
__global__ void k_deg_init(float* deg) {
  int i = blockIdx.x * 256 + threadIdx.x;
  if (i < BNN) deg[i] = 1.0f;                       // deg = scatter(1) + 1
}

__global__ void k_deg_count(const int* __restrict__ dst, float* deg) {
  int e = blockIdx.x * 256 + threadIdx.x;
  if (e < EE)
    __hip_atomic_fetch_add(&deg[dst[e]], 1.0f, __ATOMIC_RELAXED, __HIP_MEMORY_SCOPE_AGENT);
}

__global__ void k_deg_fin(float* dinv) {
  int i = blockIdx.x * 256 + threadIdx.x;
  if (i < BNN) dinv[i] = rsqrtf(dinv[i]);
}

__global__ void k_zero(float* p, int n) {
  int i = blockIdx.x * 256 + threadIdx.x;
  if (i < n) p[i] = 0.0f;
}

// ---------------- WMMA GEMM 1: m = X_w (BNx16) @ W1 (16x64) ----------------
// 1 wave -> 16x64 tile. fp32 tensor path: v_wmma_f32_16x16x4_f32.
__global__ __launch_bounds__(128) void k_gemm1(const float* __restrict__ x,
                                               const float* __restrict__ Wm,
                                               float* __restrict__ m, int w) {
  const int wave = threadIdx.x >> 5, lane = threadIdx.x & 31;
  const int tile = blockIdx.x * 4 + wave;           // 0..2499
  const int bn0  = tile * 16;
  const int b    = bn0 / NN;                        // NN % 16 == 0 -> tile in one batch
  const int n0   = bn0 - b * NN;
  const int r  = lane & 15;                         // M (A) / N (B,C)
  const int hh = lane >> 4;                         // K-half selector
  const float* arow = x + ((size_t)(b * WW + w) * NN + (n0 + r)) * 16;
  v8f acc[4] = {};
#pragma unroll
  for (int ks = 0; ks < 4; ++ks) {
    const int kb = ks * 4 + 2 * hh;                 // lane holds K = kb, kb+1
    v2f a = { arow[kb], arow[kb + 1] };
#pragma unroll
    for (int nt = 0; nt < 4; ++nt) {
      const int col = nt * 16 + r;
      v2f bb = { Wm[kb * HID + col], Wm[(kb + 1) * HID + col] };
      acc[nt] = __builtin_amdgcn_wmma_f32_16x16x4_f32(
          false, a, false, bb, (short)0, acc[nt], false, false);
    }
  }
#pragma unroll
  for (int nt = 0; nt < 4; ++nt)
#pragma unroll
    for (int v = 0; v < 8; ++v)
      m[(size_t)(bn0 + v + 8 * hh) * HID + nt * 16 + r] = acc[nt][v];
}

// ---------------- WMMA GEMM 2: m = H1 (BNx64) @ W2 (64x64) ----------------
__global__ __launch_bounds__(128) void k_gemm2(const float* __restrict__ hsrc,
                                               const float* __restrict__ Wm,
                                               float* __restrict__ m) {
  const int wave = threadIdx.x >> 5, lane = threadIdx.x & 31;
  const int tile = blockIdx.x * 4 + wave;
  const int bn0  = tile * 16;
  const int r  = lane & 15;
  const int hh = lane >> 4;
  const float* arow = hsrc + (size_t)(bn0 + r) * HID;
  v8f acc[4] = {};
#pragma unroll
  for (int ks = 0; ks < 16; ++ks) {
    const int kb = ks * 4 + 2 * hh;
    v2f a = { arow[kb], arow[kb + 1] };
#pragma unroll
    for (int nt = 0; nt < 4; ++nt) {
      const int col = nt * 16 + r;
      v2f bb = { Wm[kb * HID + col], Wm[(kb + 1) * HID + col] };
      acc[nt] = __builtin_amdgcn_wmma_f32_16x16x4_f32(
          false, a, false, bb, (short)0, acc[nt], false, false);
    }
  }
#pragma unroll
  for (int nt = 0; nt < 4; ++nt)
#pragma unroll
    for (int v = 0; v < 8; ++v)
      m[(size_t)(bn0 + v + 8 * hh) * HID + nt * 16 + r] = acc[nt][v];
}

// ---------------- edge scatter: agg[dst] += dinv[s]*dinv[d] * m[src] ----------------
// one thread = (edge, 4 features); EE*16 threads exactly.
__global__ void k_scatter(const int* __restrict__ src, const int* __restrict__ dst,
                          const float* __restrict__ dinv, const float* __restrict__ m,
                          float* __restrict__ agg) {
  int tid = blockIdx.x * 256 + threadIdx.x;
  int e = tid >> 4, f4 = (tid & 15) * 4;
  int s = src[e], d = dst[e];
  float nrm = dinv[s] * dinv[d];
  const float* ms = m + (size_t)s * HID + f4;
  float* ad = agg + (size_t)d * HID + f4;
#pragma unroll
  for (int j = 0; j < 4; ++j)
    __hip_atomic_fetch_add(&ad[j], nrm * ms[j], __ATOMIC_RELAXED, __HIP_MEMORY_SCOPE_AGENT);
}

// ---------------- combine: out = relu(agg + dinv^2 * m + bias) ----------------
__global__ void k_combine(const float* __restrict__ agg, const float* __restrict__ m,
                          const float* __restrict__ dinv, const float* __restrict__ bias,
                          float* __restrict__ out) {
  int tid = blockIdx.x * 256 + threadIdx.x;        // BNN*HID threads exactly
  int bn = tid >> 6, f = tid & 63;
  float a = (bn < NN) ? agg[tid] : 0.0f;           // edges only touch rows < NN
  float dv = dinv[bn];
  float v = a + dv * dv * m[tid] + bias[f];
  out[tid] = v > 0.0f ? v : 0.0f;
}

// ---------------- head: conv1(t=10,11) -> conv2(t=11) -> fc ----------------
__global__ void k_head(const float* __restrict__ Hb, const float* __restrict__ c1w,
                       const float* __restrict__ c1b, const float* __restrict__ c2w,
                       const float* __restrict__ c2b, const float* __restrict__ fcw,
                       const float* __restrict__ fcb, float* __restrict__ out) {
  int bn = blockIdx.x * 256 + threadIdx.x;
  if (bn >= BNN) return;
  float c1[2][32];
#pragma unroll
  for (int o = 0; o < 32; ++o) { c1[0][o] = c1b[o]; c1[1][o] = c1b[o]; }
  // Hb holds times 9,10,11 at slots 0,1,2. conv1@10 uses k=tt; conv1@11 uses k=tt-1.
  for (int tt = 0; tt < 3; ++tt) {
    const float* hp = Hb + ((size_t)tt * BNN + bn) * HID;
    const int k0 = tt;          // valid 0..2
    const int k1 = tt - 1;      // valid when >= 0 (t=12 is zero pad, never read)
    for (int i = 0; i < HID; ++i) {
      float hv = hp[i];
      const float* wrow = c1w + i * 3;
#pragma unroll
      for (int o = 0; o < 32; ++o) {
        c1[0][o] += hv * wrow[o * 192 + k0];
        if (k1 >= 0) c1[1][o] += hv * wrow[o * 192 + k1];
      }
    }
  }
#pragma unroll
  for (int t = 0; t < 2; ++t)
#pragma unroll
    for (int o = 0; o < 32; ++o)
      c1[t][o] = c1[t][o] > 0.0f ? c1[t][o] : 0.0f;

  float res = fcb[0];
#pragma unroll
  for (int o = 0; o < 32; ++o) {
    float acc = c2b[o];
    for (int j = 0; j < 32; ++j)
      acc += c1[0][j] * c2w[(o * 32 + j) * 3 + 0]
           + c1[1][j] * c2w[(o * 32 + j) * 3 + 1];   // k=2 hits pad -> dropped
    acc = acc > 0.0f ? acc : 0.0f;
    res += acc * fcw[o];
  }
  out[bn] = res;
}

extern "C" void kernel_launch(void* const* d_in, const int* in_sizes, int n_in,
                              void* d_out, int out_size, void* d_ws, size_t ws_size,
                              hipStream_t stream) {
  (void)in_sizes; (void)n_in; (void)out_size; (void)ws_size;
  const float* x   = (const float*)d_in[0];
  const int*   ei  = (const int*)  d_in[1];
  const float* W1  = (const float*)d_in[2];
  const float* b1  = (const float*)d_in[3];
  const float* W2  = (const float*)d_in[4];
  const float* b2  = (const float*)d_in[5];
  const float* c1w = (const float*)d_in[6];
  const float* c1b = (const float*)d_in[7];
  const float* c2w = (const float*)d_in[8];
  const float* c2b = (const float*)d_in[9];
  const float* fcw = (const float*)d_in[10];
  const float* fcb = (const float*)d_in[11];
  float* out = (float*)d_out;
  const int* src = ei;
  const int* dst = ei + EE;

  float* ws     = (float*)d_ws;
  float* dinv   = ws;                                  // BNN
  float* mbuf   = dinv + BNN;                          // BNN*HID
  float* aggbuf = mbuf + (size_t)BNN * HID;            // NN*HID
  float* hbuf   = aggbuf + (size_t)NN * HID;           // BNN*HID
  float* Hbuf   = hbuf + (size_t)BNN * HID;            // 3*BNN*HID

  k_deg_init<<<(BNN + 255) / 256, 256, 0, stream>>>(dinv);
  k_deg_count<<<EE / 256, 256, 0, stream>>>(dst, dinv);
  k_deg_fin<<<(BNN + 255) / 256, 256, 0, stream>>>(dinv);

  for (int wi = 0; wi < 3; ++wi) {                     // only live windows 9,10,11
    const int w = 9 + wi;
    k_gemm1<<<625, 128, 0, stream>>>(x, W1, mbuf, w);
    k_zero<<<(NN * HID) / 256, 256, 0, stream>>>(aggbuf, NN * HID);
    k_scatter<<<(EE * 16) / 256, 256, 0, stream>>>(src, dst, dinv, mbuf, aggbuf);
    k_combine<<<((size_t)BNN * HID) / 256, 256, 0, stream>>>(aggbuf, mbuf, dinv, b1, hbuf);
    k_gemm2<<<625, 128, 0, stream>>>(hbuf, W2, mbuf);
    k_zero<<<(NN * HID) / 256, 256, 0, stream>>>(aggbuf, NN * HID);
    k_scatter<<<(EE * 16) / 256, 256, 0, stream>>>(src, dst, dinv, mbuf, aggbuf);
    k_combine<<<((size_t)BNN * HID) / 256, 256, 0, stream>>>(aggbuf, mbuf, dinv, b2,
                                                             Hbuf + (size_t)wi * BNN * HID);
  }
  k_head<<<(BNN + 255) / 256, 256, 0, stream>>>(Hbuf, c1w, c1b, c2w, c2b, fcw, fcb, out);
}